// RBF_36404142801269
// MI455X (gfx1250) — compile-verified
//
#include <hip/hip_runtime.h>

// CDNA5 / gfx1250 RBF kernel map:
//   out[n,m,0] = exp(-||x_cur[n]-c_m||^2 / beta)
//   out[n,m,1] = exp(-||x_del[n]-c_m||^2 / beta)
// Distance via norm expansion; cross term 2*x·c computed with
// V_WMMA_F32_16X16X4_F32 (K=2 zero-padded to 4). Store-bandwidth bound
// (256 MB out @ 23.3 TB/s ~ 11.5 us); epilogue uses v_exp_f32 and
// non-temporal 8B stores (output > 192 MB L2, zero reuse).

typedef __attribute__((ext_vector_type(2))) float v2f;
typedef __attribute__((ext_vector_type(8))) float v8f;

#define BETA            0.04f
#define KEXP            (1.4426950408889634f / BETA)  // log2(e)/beta
#define MT              8                              // 16-wide m-tiles per wave
#define WAVES_PER_BLOCK 8
#define BLOCK_THREADS   (WAVES_PER_BLOCK * 32)

__global__ __launch_bounds__(BLOCK_THREADS) void rbf_wmma_f32_kernel(
    const float* __restrict__ xc,    // (N,2) x_cur
    const float* __restrict__ xd,    // (N,2) x_delayed
    const float* __restrict__ cen,   // (M,2) centres
    float* __restrict__ out,         // (N,M,2)
    int N, int M)
{
    const int lane  = threadIdx.x & 31;
    const int wib   = threadIdx.x >> 5;
    const int wave  = blockIdx.x * WAVES_PER_BLOCK + wib;

    const int m_chunks = M / (16 * MT);   // 32 for M=4096
    const int n_tiles  = N / 16;          // 512 for N=8192
    if (wave >= n_tiles * m_chunks) return;   // wave-uniform: EXEC stays all-1s

    const int nt     = wave / m_chunks;
    const int mc     = wave % m_chunks;
    const int n0     = nt * 16;
    const int m_base = mc * (16 * MT);

    const int row = lane & 15;   // A row / B col within tile
    const int hi  = lane >> 4;   // 0: K=0,1  1: K=2,3 (zero pad)

    const v2f zero2 = {0.f, 0.f};

    // ---- A operands: x rows, K padded 2->4 (upper half-wave holds zeros) ----
    v2f xr_a = *(const v2f*)(xc + 2 * (n0 + row));
    v2f dr_a = *(const v2f*)(xd + 2 * (n0 + row));
    v2f a_now = (hi == 0) ? xr_a : zero2;
    v2f a_del = (hi == 0) ? dr_a : zero2;

    // ---- per-lane row norms for this lane's 8 output rows (M = v + 8*hi) ----
    float xxc[8], xxd[8];
#pragma unroll
    for (int v = 0; v < 8; ++v) {
        int n = n0 + v + 8 * hi;
        v2f xr = *(const v2f*)(xc + 2 * n);
        v2f dr = *(const v2f*)(xd + 2 * n);
        xxc[v] = xr.x * xr.x + xr.y * xr.y;
        xxd[v] = dr.x * dr.x + dr.y * dr.y;
    }

#pragma unroll 1
    for (int t = 0; t < MT; ++t) {
        const int m0  = m_base + t * 16;
        const int col = m0 + row;

        // B operand: centre rows (shared by both WMMAs); also col norm
        v2f crow = *(const v2f*)(cen + 2 * col);
        float cc = crow.x * crow.x + crow.y * crow.y;
        v2f b = (hi == 0) ? crow : zero2;

        v8f cz = {};
        // dot[n,m] = x[n]·c[m]  — one 16x16 tile per WMMA
        v8f dnow = __builtin_amdgcn_wmma_f32_16x16x4_f32(
            false, a_now, false, b, (short)0, cz, false, false);
        v8f ddel = __builtin_amdgcn_wmma_f32_16x16x4_f32(
            false, a_del, false, b, (short)0, cz, false, false);

#pragma unroll
        for (int v = 0; v < 8; ++v) {
            // exp(-(xx - 2dot + cc)/beta) = exp2((2dot - xx - cc)*log2e/beta)
            float kn = __builtin_amdgcn_exp2f((2.f * dnow[v] - xxc[v] - cc) * KEXP);
            float kd = __builtin_amdgcn_exp2f((2.f * ddel[v] - xxd[v] - cc) * KEXP);
            int n = n0 + v + 8 * hi;
            v2f o = {kn, kd};
            // coalesced 8B NT store: lanes 0-15 and 16-31 each write a
            // contiguous 128B row segment of out[n, m0:m0+16, :]
            __builtin_nontemporal_store(o, (v2f*)(out + 2 * ((size_t)n * M + col)));
        }
    }
}

extern "C" void kernel_launch(void* const* d_in, const int* in_sizes, int n_in,
                              void* d_out, int out_size, void* d_ws, size_t ws_size,
                              hipStream_t stream) {
    const float* xc  = (const float*)d_in[0];   // x_cur     (N,2)
    const float* xd  = (const float*)d_in[1];   // x_delayed (N,2)
    const float* cen = (const float*)d_in[2];   // centres   (M,2)
    float* out = (float*)d_out;                 // (N,M,2)

    const int N = in_sizes[0] / 2;
    const int M = in_sizes[2] / 2;

    const int m_chunks = M / (16 * MT);
    const int n_tiles  = N / 16;
    const int waves    = n_tiles * m_chunks;
    const int blocks   = (waves + WAVES_PER_BLOCK - 1) / WAVES_PER_BLOCK;

    rbf_wmma_f32_kernel<<<blocks, BLOCK_THREADS, 0, stream>>>(xc, xd, cen, out, N, M);
}